// RolandGNN_1614907703850
// MI455X (gfx1250) — compile-verified
//
#include <hip/hip_runtime.h>
#include <cstddef>

typedef float v2f __attribute__((ext_vector_type(2)));
typedef float v8f __attribute__((ext_vector_type(8)));

constexpr int   NODES = 50000;
constexpr int   EDGES = 800000;
constexpr int   HID   = 128;
constexpr float NEG   = 0.01f;

constexpr int KK        = 128;       // contraction dim (always 128 here)
constexpr int BT_STRIDE = KK + 2;    // LDS pad: avoids bank conflicts, keeps b64 align

// ---------------------------------------------------------------------------
// fp32 WMMA GEMM: C[M,N] = act(A[M,KK] @ B[KK,N] + bias)
// Block = 256 threads (8 waves). All 8 waves share one 64-column n-group;
// the B panel (KK x 64) is staged transposed into LDS once per block so each
// B fragment is a single ds_load_b64. Each wave owns one 16-row m-tile.
// act: 0 = identity, 1 = leaky relu(0.01). bias may be nullptr.
// Requires M % 16 == 0, N % 64 == 0.
// ---------------------------------------------------------------------------
__global__ __launch_bounds__(256) void k_gemm(const float* __restrict__ A,
                                              const float* __restrict__ B,
                                              const float* __restrict__ bias,
                                              float* __restrict__ C,
                                              int M, int N, int act)
{
    __shared__ __align__(16) float bt[64 * BT_STRIDE];   // bt[c][k] = B[k][n0+c]

    const int tid  = threadIdx.x;
    const int lane = tid & 31;
    const int wave = tid >> 5;
    const int n0   = blockIdx.y << 6;

    // Stage B panel transposed: coalesced global reads, padded LDS writes.
    for (int idx = tid; idx < 64 * KK; idx += 256) {
        const int k = idx >> 6;
        const int c = idx & 63;
        bt[c * BT_STRIDE + k] = B[(size_t)k * N + n0 + c];
    }
    __syncthreads();

    // Overhang waves clamp to the last m-tile: duplicate work, identical stores
    // (keeps EXEC all-1s through the WMMA loop and past the barrier).
    int mtile = blockIdx.x * 8 + wave;
    const int mtiles = M >> 4;
    if (mtile >= mtiles) mtile = mtiles - 1;
    const int m0   = mtile << 4;
    const int half = lane >> 4;                  // 0: lanes 0-15, 1: lanes 16-31
    const int l16  = lane & 15;

    v8f acc0 = {}, acc1 = {}, acc2 = {}, acc3 = {};

    // A fragment (16x4 f32): lanes 0-15 hold K = k0,k0+1 ; lanes 16-31 hold k0+2,k0+3
    const float* arow = A + (size_t)(m0 + l16) * KK + 2 * half;
    // B fragment base in LDS: column l16 (+16/32/48 per n-tile), rows k0+2*half, +1
    const float* btp = &bt[l16 * BT_STRIDE + 2 * half];

    for (int k0 = 0; k0 < KK; k0 += 4) {
        v2f a;
        a.x = arow[k0];
        a.y = arow[k0 + 1];

        const v2f b0 = *(const v2f*)(btp + k0);
        const v2f b1 = *(const v2f*)(btp + 16 * BT_STRIDE + k0);
        const v2f b2 = *(const v2f*)(btp + 32 * BT_STRIDE + k0);
        const v2f b3 = *(const v2f*)(btp + 48 * BT_STRIDE + k0);

        acc0 = __builtin_amdgcn_wmma_f32_16x16x4_f32(false, a, false, b0, (short)0, acc0, false, false);
        acc1 = __builtin_amdgcn_wmma_f32_16x16x4_f32(false, a, false, b1, (short)0, acc1, false, false);
        acc2 = __builtin_amdgcn_wmma_f32_16x16x4_f32(false, a, false, b2, (short)0, acc2, false, false);
        acc3 = __builtin_amdgcn_wmma_f32_16x16x4_f32(false, a, false, b3, (short)0, acc3, false, false);
    }

    // C/D layout: VGPR r -> row m0+r (lanes 0-15) / m0+r+8 (lanes 16-31)
    v8f accs[4] = {acc0, acc1, acc2, acc3};
    #pragma unroll
    for (int t = 0; t < 4; ++t) {
        const int   n  = n0 + t * 16 + l16;
        const float bv = bias ? bias[n] : 0.0f;
        #pragma unroll
        for (int r = 0; r < 8; ++r) {
            const int row = m0 + r + half * 8;
            float v = accs[t][r] + bv;
            if (act) v = (v >= 0.0f) ? v : NEG * v;
            C[(size_t)row * N + n] = v;
        }
    }
}

// ---------------------------------------------------------------------------
// Small helper kernels
// ---------------------------------------------------------------------------
__global__ void k_fill(float* __restrict__ p, float v, int n)
{
    int i = blockIdx.x * blockDim.x + threadIdx.x;
    if (i < n) p[i] = v;
}

__global__ void k_deg(const long long* __restrict__ ei, float* __restrict__ deg)
{
    int e = blockIdx.x * blockDim.x + threadIdx.x;
    if (e < EDGES) atomicAdd(&deg[(int)ei[EDGES + e]], 1.0f);
}

__global__ void k_rsqrt(float* __restrict__ p, int n)
{
    int i = blockIdx.x * blockDim.x + threadIdx.x;
    if (i < n) p[i] = rsqrtf(p[i]);
}

// One wave per edge; lane covers 4 consecutive features of the 128-wide row.
__global__ __launch_bounds__(256) void k_scatter(const long long* __restrict__ ei,
                                                 const float* __restrict__ hw,
                                                 const float* __restrict__ dinv,
                                                 float* __restrict__ agg)
{
    const int e    = blockIdx.x * 8 + (threadIdx.x >> 5);
    const int lane = threadIdx.x & 31;
    if (e >= EDGES) return;
    const int   s    = (int)ei[e];
    const int   d    = (int)ei[EDGES + e];
    const float coef = dinv[s] * dinv[d];
    const float4 v = ((const float4*)(hw + (size_t)s * HID))[lane];
    float* ap = agg + (size_t)d * HID + lane * 4;
    atomicAdd(ap + 0, v.x * coef);
    atomicAdd(ap + 1, v.y * coef);
    atomicAdd(ap + 2, v.z * coef);
    atomicAdd(ap + 3, v.w * coef);
}

// h = lrelu(agg + hw * dinv[n]^2 + b[f]) written in place over hw.
__global__ void k_gcn_fin(const float* __restrict__ agg,
                          const float* __restrict__ dinv,
                          const float* __restrict__ b,
                          float* __restrict__ hw_io)
{
    int i = blockIdx.x * blockDim.x + threadIdx.x;
    if (i >= NODES * HID) return;
    const int   n  = i >> 7;          // / HID
    const int   f  = i & (HID - 1);
    const float di = dinv[n];
    float v = agg[i] + hw_io[i] * di * di + b[f];
    hw_io[i] = (v >= 0.0f) ? v : NEG * v;
}

__device__ __forceinline__ float sigmoidf(float x) { return 1.0f / (1.0f + expf(-x)); }

__global__ void k_gru(const float* __restrict__ gi, const float* __restrict__ gh,
                      const float* __restrict__ prev, float* __restrict__ out)
{
    int i = blockIdx.x * blockDim.x + threadIdx.x;
    if (i >= NODES * HID) return;
    const int n = i >> 7;
    const int f = i & (HID - 1);
    const size_t base = (size_t)n * 3 * HID;
    const float r  = sigmoidf(gi[base + f]           + gh[base + f]);
    const float z  = sigmoidf(gi[base + HID + f]     + gh[base + HID + f]);
    const float nn = tanhf   (gi[base + 2 * HID + f] + r * gh[base + 2 * HID + f]);
    out[i] = (1.0f - z) * nn + z * prev[i];
}

__global__ void k_post(const float* __restrict__ h, const float* __restrict__ w,
                       const float* __restrict__ b, float* __restrict__ out)
{
    int n = blockIdx.x * blockDim.x + threadIdx.x;
    if (n >= NODES) return;
    const float* hp = h + (size_t)n * HID;
    float acc = 0.0f;
    #pragma unroll 4
    for (int f = 0; f < HID; ++f) acc += hp[f] * w[f];
    out[n] = acc + b[0];
}

// ---------------------------------------------------------------------------
extern "C" void kernel_launch(void* const* d_in, const int* in_sizes, int n_in,
                              void* d_out, int out_size, void* d_ws, size_t ws_size,
                              hipStream_t stream)
{
    (void)in_sizes; (void)n_in; (void)out_size; (void)ws_size;

    const float*     x      = (const float*)d_in[0];
    const long long* ei     = (const long long*)d_in[1];   // int64 [2, EDGES]
    const float*     w_pre1 = (const float*)d_in[2];
    const float*     b_pre1 = (const float*)d_in[3];
    const float*     w_pre2 = (const float*)d_in[4];
    const float*     b_pre2 = (const float*)d_in[5];
    const float*     w_gcn[2] = {(const float*)d_in[6],  (const float*)d_in[8]};
    const float*     b_gcn[2] = {(const float*)d_in[7],  (const float*)d_in[9]};
    const float*     wih[2]   = {(const float*)d_in[10], (const float*)d_in[14]};
    const float*     whh[2]   = {(const float*)d_in[11], (const float*)d_in[15]};
    const float*     bih[2]   = {(const float*)d_in[12], (const float*)d_in[16]};
    const float*     bhh[2]   = {(const float*)d_in[13], (const float*)d_in[17]};
    const float*     prev[2]  = {(const float*)d_in[18], (const float*)d_in[19]};
    const float*     w_post = (const float*)d_in[20];
    const float*     b_post = (const float*)d_in[21];

    // Output layout: out[NODES] | emb0[NODES*HID] | emb1[NODES*HID]
    float* out  = (float*)d_out;
    float* emb0 = out + NODES;
    float* emb1 = emb0 + (size_t)NODES * HID;
    float* embs[2] = {emb0, emb1};

    // Workspace layout (agg aliased onto gi: their lifetimes do not overlap)
    float* ws   = (float*)d_ws;
    float* bufA = ws;                                  // [NODES*HID]  hw / gcn-out
    float* bufB = bufA + (size_t)NODES * HID;          // [NODES*HID]  pre-MLP h
    float* gi   = bufB + (size_t)NODES * HID;          // [NODES*3*HID]
    float* gh   = gi   + (size_t)NODES * 3 * HID;      // [NODES*3*HID]
    float* dinv = gh   + (size_t)NODES * 3 * HID;      // [NODES]
    float* agg  = gi;

    const int TB = 256;
    const dim3 blk(TB);
    const dim3 gN((NODES + TB - 1) / TB);
    const dim3 gE((EDGES + TB - 1) / TB);
    const dim3 gNH((NODES * HID + TB - 1) / TB);
    const dim3 gemm128((3125 + 7) / 8, 128 / 64);      // 50000/16 = 3125 m-tiles
    const dim3 gemm384((3125 + 7) / 8, 384 / 64);

    // Degree + symmetric normalization (graph-only, shared by both layers)
    k_fill<<<gN, blk, 0, stream>>>(dinv, 1.0f, NODES);          // self-loop
    k_deg<<<gE, blk, 0, stream>>>(ei, dinv);
    k_rsqrt<<<gN, blk, 0, stream>>>(dinv, NODES);

    // Pre-MLP: h = lrelu(lrelu(x W1 + b1) W2 + b2)
    k_gemm<<<gemm128, blk, 0, stream>>>(x,    w_pre1, b_pre1, bufA, NODES, HID, 1);
    k_gemm<<<gemm128, blk, 0, stream>>>(bufA, w_pre2, b_pre2, bufB, NODES, HID, 1);

    const float* h = bufB;
    for (int L = 0; L < 2; ++L) {
        // hw = h @ Wg (no bias/act)
        k_gemm<<<gemm128, blk, 0, stream>>>(h, w_gcn[L], nullptr, bufA, NODES, HID, 0);
        // agg = segment_sum(norm * hw[src], dst)
        k_fill<<<gNH, blk, 0, stream>>>(agg, 0.0f, NODES * HID);
        k_scatter<<<dim3(EDGES / 8), blk, 0, stream>>>(ei, bufA, dinv, agg);
        // gcn-out = lrelu(agg + hw*dinv^2 + b)   (in place over bufA)
        k_gcn_fin<<<gNH, blk, 0, stream>>>(agg, dinv, b_gcn[L], bufA);
        // GRU gate pre-activations (agg/gi alias is safe: agg is dead here)
        k_gemm<<<gemm384, blk, 0, stream>>>(bufA,    wih[L], bih[L], gi, NODES, 3 * HID, 0);
        k_gemm<<<gemm384, blk, 0, stream>>>(prev[L], whh[L], bhh[L], gh, NODES, 3 * HID, 0);
        // GRU combine -> emb_L (also next layer's h)
        k_gru<<<gNH, blk, 0, stream>>>(gi, gh, prev[L], embs[L]);
        h = embs[L];
    }

    // out = emb1 @ w_post + b_post
    k_post<<<gN, blk, 0, stream>>>(emb1, w_post, b_post, out);
}